// DiceLossInt_67250597920980
// MI455X (gfx1250) — compile-verified
//
#include <hip/hip_runtime.h>

// ---- problem constants (match reference) ----
#define NUM_CLASSES 32
#define NBATCH      4
#define NPB         (256*256*256)   // elements per batch = 16,777,216
#define WGPB        256             // workgroups per batch
#define BLOCK       256
#define ELEMS_PER_WG (NPB / WGPB)   // 65536
#define VECS_PER_WG  (ELEMS_PER_WG / 4) // 16384

typedef int   v4i __attribute__((ext_vector_type(4)));
typedef float v2f __attribute__((ext_vector_type(2)));
typedef float v8f __attribute__((ext_vector_type(8)));

// ---------------------------------------------------------------------------
// Kernel 0: zero the global joint-histogram workspace (4 x 1024 u32)
// ---------------------------------------------------------------------------
__global__ void dice_zero_kernel(unsigned* __restrict__ gH2) {
    int i = blockIdx.x * BLOCK + threadIdx.x;
    if (i < NBATCH * 1024) gH2[i] = 0u;
}

// ---------------------------------------------------------------------------
// Kernel 1: streaming joint histogram.
//   key = in*32 + tg  -> 1024-bin LDS histogram per workgroup (1 ds_add/pair),
//   then merged into gH2[b][1024] with global atomics.
// ---------------------------------------------------------------------------
__global__ void __launch_bounds__(BLOCK)
dice_hist_kernel(const int* __restrict__ in, const int* __restrict__ tg,
                 unsigned* __restrict__ gH2) {
    __shared__ unsigned h2[1024];
    const int t = threadIdx.x;
    #pragma unroll
    for (int i = t; i < 1024; i += BLOCK) h2[i] = 0u;
    __syncthreads();

    const int b = blockIdx.y;
    const long long base = (long long)b * NPB + (long long)blockIdx.x * ELEMS_PER_WG;
    const v4i* __restrict__ inp = (const v4i*)(in + base);
    const v4i* __restrict__ tgp = (const v4i*)(tg + base);

    for (int v = t; v < VECS_PER_WG; v += BLOCK) {
        v4i a = __builtin_nontemporal_load(inp + v);
        v4i c = __builtin_nontemporal_load(tgp + v);
        // prefetch next iteration's cachelines (global_prefetch_b8)
        if (v + BLOCK < VECS_PER_WG) {
            __builtin_prefetch((const void*)(inp + v + BLOCK), 0, 0);
            __builtin_prefetch((const void*)(tgp + v + BLOCK), 0, 0);
        }
        #pragma unroll
        for (int j = 0; j < 4; ++j) {
            unsigned key = ((unsigned)a[j] << 5) | (unsigned)c[j]; // in*32 + tg
            atomicAdd(&h2[key], 1u);
        }
    }
    __syncthreads();

    #pragma unroll
    for (int i = t; i < 1024; i += BLOCK) {
        unsigned cnt = h2[i];
        if (cnt) atomicAdd(&gH2[b * 1024 + i], cnt);
    }
}

// ---------------------------------------------------------------------------
// Kernel 2: finalization (1 wave of 32 threads).
//   For each batch:  S = H2 + H2^T ;  total[c] = rowsum(S)[c] via chained
//   V_WMMA_F32_16X16X4_F32 with B = ones (layout-invariant);
//   inter[c] = H2[c][c];  dice = 1 - (2I+s)/(T+s); sum classes, mean batches.
//   Counts < 2^24 so f32 arithmetic is exact for the histogram values.
// ---------------------------------------------------------------------------
__global__ void dice_final_kernel(const unsigned* __restrict__ gH2,
                                  const float* __restrict__ smooth_p,
                                  float* __restrict__ out) {
    const int lane = threadIdx.x;            // exactly 32 threads -> full wave32
    __shared__ float rs[32];
    __shared__ float lsum[32];

    const float s = *smooth_p;
    float total_loss = 0.0f;

    v2f ones2;
    ones2.x = 1.0f;
    ones2.y = 1.0f;                          // B (4x16) = all ones, any layout

    const int mrow = lane & 15;              // M index within 16-row block
    const int koff = (lane >> 4) << 1;       // lanes 0-15 -> K+0, lanes 16-31 -> K+2

    for (int b = 0; b < NBATCH; ++b) {
        const unsigned* __restrict__ H = gH2 + b * 1024;

        #pragma unroll
        for (int rb = 0; rb < 2; ++rb) {     // row blocks: classes [0,16) and [16,32)
            v8f acc = {};
            const int m = rb * 16 + mrow;
            #pragma unroll
            for (int kb = 0; kb < 8; ++kb) { // K = 32 in chunks of 4
                const int k0 = kb * 4 + koff;
                // A(16x4) slice of S = H + H^T, per ISA 32-bit A layout
                v2f A;
                A.x = (float)H[m * 32 + k0]     + (float)H[k0 * 32 + m];
                A.y = (float)H[m * 32 + k0 + 1] + (float)H[(k0 + 1) * 32 + m];
                acc = __builtin_amdgcn_wmma_f32_16x16x4_f32(
                        false, A, false, ones2, (short)0, acc, false, false);
            }
            // C/D layout: lane<16: acc[j] = rowsum[rb*16+j]; lane>=16: rowsum[rb*16+8+j]
            if (lane == 0) {
                #pragma unroll
                for (int j = 0; j < 8; ++j) rs[rb * 16 + j] = acc[j];
            } else if (lane == 16) {
                #pragma unroll
                for (int j = 0; j < 8; ++j) rs[rb * 16 + 8 + j] = acc[j];
            }
        }
        __syncthreads();

        // lane c handles class c
        const float inter = (float)H[lane * 33];     // diagonal H2[c][c]
        const float total = rs[lane];                // hist_in[c] + hist_tg[c]
        const float dice  = 1.0f - (2.0f * inter + s) / (total + s);
        lsum[lane] = dice;
        __syncthreads();

        if (lane == 0) {
            float t = 0.0f;
            #pragma unroll
            for (int i = 0; i < 32; ++i) t += lsum[i];
            total_loss += t;
        }
        __syncthreads();
    }

    if (lane == 0) out[0] = total_loss * (1.0f / (float)NBATCH);
}

// ---------------------------------------------------------------------------
extern "C" void kernel_launch(void* const* d_in, const int* in_sizes, int n_in,
                              void* d_out, int out_size, void* d_ws, size_t ws_size,
                              hipStream_t stream) {
    (void)in_sizes; (void)n_in; (void)out_size; (void)ws_size;
    const int*   inputs  = (const int*)d_in[0];
    const int*   targets = (const int*)d_in[1];
    const float* smooth  = (const float*)d_in[2];
    unsigned*    gH2     = (unsigned*)d_ws;      // NBATCH * 1024 u32 = 16 KB
    float*       out     = (float*)d_out;

    dice_zero_kernel<<<(NBATCH * 1024 + BLOCK - 1) / BLOCK, BLOCK, 0, stream>>>(gH2);
    dice_hist_kernel<<<dim3(WGPB, NBATCH), BLOCK, 0, stream>>>(inputs, targets, gH2);
    dice_final_kernel<<<1, 32, 0, stream>>>(gH2, smooth, out);
}